// CSOCSSC_v42_58497454571796
// MI455X (gfx1250) — compile-verified
//
#include <hip/hip_runtime.h>
#include <hip/hip_bf16.h>
#include <math.h>

typedef __attribute__((ext_vector_type(2))) float v2f;
typedef __attribute__((ext_vector_type(4))) float v4f;
typedef __attribute__((ext_vector_type(8))) float v8f;

#define CLAMP_D 10.0f
#define EPS_NORM 1e-8f
#define WAVES_PER_BLOCK 8
#define TILE 16
#define JTILES 4   // j-tiles swept per wave with one A fragment

// ---------------------------------------------------------------------------
// Kernel 1: per-residue prep.
//   diff^2_ij = e_j^T (R_i R_i^T) e_j = dot(m6_i, q6_j) over 6 logical K's.
//   Logical K order: [K0..K5] = [M00, M11, M22, 2M01, 2M02, 2M12]  (A side)
//                               [ex^2, ey^2, ez^2, ex ey, ex ez, ey ez] (B side)
//   Physical store order is permuted for one-b128-per-lane-half WMMA loads:
//     phys[0..3] = K0,K1,K4,K5   (lane half 0: WMMA1 pair + WMMA2 pair)
//     phys[4..7] = K2,K3, 0, 0   (lane half 1)
//   Masks are folded in multiplicatively: masked row/col -> d2 = 0 -> pp = 0.
// ---------------------------------------------------------------------------
__global__ void fape_prep_kernel(const float* __restrict__ n,
                                 const float* __restrict__ ca,
                                 const float* __restrict__ c,
                                 const float* __restrict__ pred,
                                 const float* __restrict__ truep,
                                 const unsigned char* __restrict__ mask,
                                 float* __restrict__ m6,
                                 float* __restrict__ q6,
                                 float* __restrict__ maskf,
                                 int total) {
  int idx = blockIdx.x * blockDim.x + threadIdx.x;
  if (idx >= total) return;
  int base = idx * 3;

  float cax = ca[base + 0], cay = ca[base + 1], caz = ca[base + 2];

  // x = normalize(c - ca)
  float xx = c[base + 0] - cax, xy = c[base + 1] - cay, xz = c[base + 2] - caz;
  float inv = 1.0f / (sqrtf(xx * xx + xy * xy + xz * xz) + EPS_NORM);
  xx *= inv; xy *= inv; xz *= inv;

  // v = normalize(n - ca)
  float vx = n[base + 0] - cax, vy = n[base + 1] - cay, vz = n[base + 2] - caz;
  inv = 1.0f / (sqrtf(vx * vx + vy * vy + vz * vz) + EPS_NORM);
  vx *= inv; vy *= inv; vz *= inv;

  // z = normalize(cross(x, v))
  float zx = xy * vz - xz * vy;
  float zy = xz * vx - xx * vz;
  float zz = xx * vy - xy * vx;
  inv = 1.0f / (sqrtf(zx * zx + zy * zy + zz * zz) + EPS_NORM);
  zx *= inv; zy *= inv; zz *= inv;

  // y = normalize(cross(z, x))
  float yx = zy * xz - zz * xy;
  float yy = zz * xx - zx * xz;
  float yz = zx * xy - zy * xx;
  inv = 1.0f / (sqrtf(yx * yx + yy * yy + yz * yz) + EPS_NORM);
  yx *= inv; yy *= inv; yz *= inv;

  // M = x x^T + y y^T + z z^T  (Gram of frame columns)
  float M00 = xx * xx + yx * yx + zx * zx;
  float M11 = xy * xy + yy * yy + zy * zy;
  float M22 = xz * xz + yz * yz + zz * zz;
  float M01 = xx * xy + yx * yy + zx * zy;
  float M02 = xx * xz + yx * yz + zx * zz;
  float M12 = xy * xz + yy * yz + zy * zz;

  const float mk = mask[idx] ? 1.0f : 0.0f;

  // phys order: [K0,K1,K4,K5, K2,K3,0,0]
  float* mo = m6 + (size_t)idx * 8;
  mo[0] = mk * M00;          // K0
  mo[1] = mk * M11;          // K1
  mo[2] = mk * 2.0f * M02;   // K4
  mo[3] = mk * 2.0f * M12;   // K5
  mo[4] = mk * M22;          // K2
  mo[5] = mk * 2.0f * M01;   // K3
  mo[6] = 0.0f;
  mo[7] = 0.0f;

  float ex = pred[base + 0] - truep[base + 0];
  float ey = pred[base + 1] - truep[base + 1];
  float ez = pred[base + 2] - truep[base + 2];
  float* qo = q6 + (size_t)idx * 8;
  qo[0] = mk * ex * ex;      // K0
  qo[1] = mk * ey * ey;      // K1
  qo[2] = mk * ex * ez;      // K4
  qo[3] = mk * ey * ez;      // K5
  qo[4] = mk * ez * ez;      // K2
  qo[5] = mk * ex * ey;      // K3
  qo[6] = 0.0f;
  qo[7] = 0.0f;

  maskf[idx] = mk;
}

// ---------------------------------------------------------------------------
// Kernel 2: one wave32 per (16 x (JTILES*16)) strip. A fragment loaded once
// (one b128 per lane), then per j-tile: one b128 B load + two chained
// V_WMMA_F32_16X16X4_F32 (K=8, top 2 zero-padded) + raw-v_sqrt epilogue.
//
// A 16x4 f32 layout: lanes 0-15 row M=lane (K0 in VGPR0, K1 in VGPR1),
// lanes 16-31 same rows (K2/K3). Phys permutation makes each half's four
// needed scalars contiguous at row*8 + 4*half.
// C/D layout: VGPR v -> row (v + 8*half), col (lane&15).
// ---------------------------------------------------------------------------
__global__ void __launch_bounds__(WAVES_PER_BLOCK * 32)
fape_tile_kernel(const float* __restrict__ m6,
                 const float* __restrict__ q6,
                 float* __restrict__ partials,
                 int N) {
  const int lane = threadIdx.x & 31;
  const int wave = threadIdx.x >> 5;
  const int half = lane >> 4;
  const int m    = lane & 15;

  const int tilesPerRow    = N / TILE;           // 64
  const int jGroups        = tilesPerRow / JTILES;
  const int groupsPerBatch = tilesPerRow * jGroups;

  const int group = blockIdx.x * WAVES_PER_BLOCK + wave;
  const int b     = group / groupsPerBatch;
  const int rem   = group - b * groupsPerBatch;
  const int ti    = rem / jGroups;
  const int jg    = rem - ti * jGroups;
  const int i0    = ti * TILE;
  const int j0b   = jg * (JTILES * TILE);

  const v4f av = *(const v4f*)(m6 + ((size_t)(b * N + i0 + m)) * 8 + 4 * half);
  const v2f a1 = {av.x, av.y};   // K(2h), K(2h+1)
  const v2f a2 = {av.z, av.w};   // K(4+2h), K(5+2h)

  float s = 0.0f;
  #pragma unroll
  for (int t = 0; t < JTILES; ++t) {
    const int j0 = j0b + t * TILE;
    const v4f bv = *(const v4f*)(q6 + ((size_t)(b * N + j0 + m)) * 8 + 4 * half);
    const v2f b1 = {bv.x, bv.y};
    const v2f b2 = {bv.z, bv.w};

    v8f acc = {};
    // 8 args: (neg_a, A, neg_b, B, c_mod, C, reuse_a, reuse_b)
    acc = __builtin_amdgcn_wmma_f32_16x16x4_f32(false, a1, false, b1,
                                                (short)0, acc, false, false);
    acc = __builtin_amdgcn_wmma_f32_16x16x4_f32(false, a2, false, b2,
                                                (short)0, acc, false, false);

    #pragma unroll
    for (int v = 0; v < 8; ++v) {
      const float diff = __builtin_amdgcn_sqrtf(fmaxf(acc[v], 0.0f));
      s += diff + fminf(diff, CLAMP_D);   // == 2 * (clamped + 0.5*(diff-clamped))
    }
  }
  s *= 0.5f;

  __shared__ float red[WAVES_PER_BLOCK * 32];
  red[threadIdx.x] = s;
  __syncthreads();
  if (threadIdx.x == 0) {
    float t = 0.0f;
    #pragma unroll 8
    for (int k = 0; k < WAVES_PER_BLOCK * 32; ++k) t += red[k];
    partials[blockIdx.x] = t;
  }
}

// ---------------------------------------------------------------------------
// Kernel 3: single block, fixed-order reduction of partials and mask sum.
// ---------------------------------------------------------------------------
__global__ void __launch_bounds__(256)
fape_final_kernel(const float* __restrict__ partials, int nPart,
                  const float* __restrict__ maskf, int nMask,
                  float* __restrict__ out) {
  __shared__ float red[256];

  float s = 0.0f;
  for (int k = threadIdx.x; k < nPart; k += 256) s += partials[k];
  red[threadIdx.x] = s;
  __syncthreads();
  for (int off = 128; off > 0; off >>= 1) {
    if (threadIdx.x < off) red[threadIdx.x] += red[threadIdx.x + off];
    __syncthreads();
  }
  const float total = red[0];
  __syncthreads();

  float sm = 0.0f;
  for (int k = threadIdx.x; k < nMask; k += 256) sm += maskf[k];
  red[threadIdx.x] = sm;
  __syncthreads();
  for (int off = 128; off > 0; off >>= 1) {
    if (threadIdx.x < off) red[threadIdx.x] += red[threadIdx.x + off];
    __syncthreads();
  }
  if (threadIdx.x == 0) out[0] = total / (red[0] + 1e-8f);
}

// ---------------------------------------------------------------------------
extern "C" void kernel_launch(void* const* d_in, const int* in_sizes, int n_in,
                              void* d_out, int out_size, void* d_ws, size_t ws_size,
                              hipStream_t stream) {
  const float* n_ptr    = (const float*)d_in[0];
  const float* ca_ptr   = (const float*)d_in[1];
  const float* c_ptr    = (const float*)d_in[2];
  const float* pred_ptr = (const float*)d_in[3];
  const float* true_ptr = (const float*)d_in[4];
  const unsigned char* mask_ptr = (const unsigned char*)d_in[5];
  float* out = (float*)d_out;

  const int BN = in_sizes[5];      // B*N residues (mask element count)
  const int N  = 1024;
  const int B  = BN / N;

  // Workspace layout (floats)
  float* ws       = (float*)d_ws;
  float* m6       = ws;                       // BN*8
  float* q6       = m6 + (size_t)BN * 8;      // BN*8
  float* maskf    = q6 + (size_t)BN * 8;      // BN
  float* partials = maskf + BN;               // groups/WAVES_PER_BLOCK

  // Kernel 1: prep
  {
    int threads = 256;
    int blocks = (BN + threads - 1) / threads;
    fape_prep_kernel<<<blocks, threads, 0, stream>>>(
        n_ptr, ca_ptr, c_ptr, pred_ptr, true_ptr, mask_ptr,
        m6, q6, maskf, BN);
  }

  // Kernel 2: WMMA strips (one A fragment reused across JTILES j-tiles)
  const int tilesPerRow = N / TILE;                           // 64
  const int jGroups = tilesPerRow / JTILES;                   // 16
  const int groups  = B * tilesPerRow * jGroups;              // 8192
  const int blocks2 = groups / WAVES_PER_BLOCK;               // 1024
  fape_tile_kernel<<<blocks2, WAVES_PER_BLOCK * 32, 0, stream>>>(
      m6, q6, partials, N);

  // Kernel 3: final deterministic reduction
  fape_final_kernel<<<1, 256, 0, stream>>>(partials, blocks2, maskf, BN, out);
}